// GaussMonom_34772055228970
// MI455X (gfx1250) — compile-verified
//
#include <hip/hip_runtime.h>

// CDNA5 / gfx1250: wave32, WMMA f32 16x16x4.
typedef __attribute__((ext_vector_type(2))) float v2f;
typedef __attribute__((ext_vector_type(8))) float v8f;

#define BLOCK_THREADS 256

__global__ __launch_bounds__(BLOCK_THREADS) void
GaussMonom_34772055228970_kernel(const v2f*   __restrict__ pts,
                                 const float* __restrict__ mean,
                                 const float* __restrict__ cov,
                                 const float* __restrict__ cnst,
                                 float*       __restrict__ out,
                                 int n)
{
    // Uniform scalars (compiler scalarizes these loads).
    const float mx = mean[0], my = mean[1];
    const float w0 = cov[0];            // c00
    const float w1 = cov[1] + cov[2];   // c01 + c10
    const float w2 = cov[3];            // c11
    const float cst = cnst[0];

    const int lane = threadIdx.x & 31;
    const int hi   = lane & 16;         // lanes 16..31
    const int q    = lane & 15;

    const long long tid  = (long long)blockIdx.x * BLOCK_THREADS + threadIdx.x;
    const long long base = tid & ~31LL; // this wave's 32-point group

#if __has_builtin(__builtin_amdgcn_wmma_f32_16x16x4_f32)
    if (base + 32 <= (long long)n) {
        // ---- A (16x4): every row = [w0, w1, w2, 0] ----
        // ISA layout: lanes 0-15 hold K=0,1 ; lanes 16-31 hold K=2,3.
        v2f A;
        A.x = hi ? w2  : w0;
        A.y = hi ? 0.f : w1;

        // ---- B (4x16): column n = [dx^2, dx*dy, dy^2, 0] of point n ----
        // Both half-waves read the same 16 points -> one 128B request each.
        v2f p0 = __builtin_nontemporal_load(&pts[base + q]);       // group 0: base+0..15
        v2f p1 = __builtin_nontemporal_load(&pts[base + 16 + q]);  // group 1: base+16..31
        float dx0 = p0.x - mx, dy0 = p0.y - my;
        float dx1 = p1.x - mx, dy1 = p1.y - my;

        v2f B0, B1;
        { float e = hi ? dy0 : dx0; float f = hi ? 0.f : dx0; B0.x = e * e; B0.y = f * dy0; }
        { float e = hi ? dy1 : dx1; float f = hi ? 0.f : dx1; B1.x = e * e; B1.y = f * dy1; }

        // D[m,n] = w . u_n = zeta_n  (replicated across rows m)
        v8f cz = {};
        v8f D0 = __builtin_amdgcn_wmma_f32_16x16x4_f32(
                     false, A, false, B0, (short)0, cz, false, false);
        v8f D1 = __builtin_amdgcn_wmma_f32_16x16x4_f32(
                     false, A, false, B1, (short)0, cz, false, false);

        // vgpr0: lanes 0-15 -> zeta[base+lane] (D0), lanes 16-31 -> zeta[base+lane] (D1, M=8 row)
        float zeta = hi ? D1[0] : D0[0];
        // Fully coalesced 128B store across the wave; NT: output is never re-read.
        __builtin_nontemporal_store(cst * __expf(-zeta), &out[base + lane]);
        return;
    }
#endif
    // Tail group (uniform branch per wave, so WMMA above always runs with EXEC==all 1s)
    // and fallback if the f32 WMMA builtin is unavailable.
    if (tid < (long long)n) {
        v2f p = __builtin_nontemporal_load(&pts[tid]);
        float dx = p.x - mx, dy = p.y - my;
        float zeta = w0 * dx * dx + w1 * dx * dy + w2 * dy * dy;
        __builtin_nontemporal_store(cst * __expf(-zeta), &out[tid]);
    }
}

extern "C" void kernel_launch(void* const* d_in, const int* in_sizes, int n_in,
                              void* d_out, int out_size, void* d_ws, size_t ws_size,
                              hipStream_t stream) {
    (void)n_in; (void)d_ws; (void)ws_size;
    const v2f*   pts  = (const v2f*)d_in[0];      // [N,2] float32
    const float* mean = (const float*)d_in[1];    // [2]
    const float* cov  = (const float*)d_in[2];    // [2,2]
    const float* cst  = (const float*)d_in[3];    // [1]
    float*       out  = (float*)d_out;            // [N]

    const int n = in_sizes[0] / 2;
    (void)out_size;
    const int blocks = (n + BLOCK_THREADS - 1) / BLOCK_THREADS;
    GaussMonom_34772055228970_kernel<<<blocks, BLOCK_THREADS, 0, stream>>>(
        pts, mean, cov, cst, out, n);
}